// TrilinearInterpolationGS_3358664425833
// MI455X (gfx1250) — compile-verified
//
#include <hip/hip_runtime.h>

// ---------------------------------------------------------------------------
// Trilinear 3D-LUT application (color-grading LUT), MI455X / gfx1250.
//   lut: (3,33,33,33) f32  (431 KB -> fully L2 resident; 192MB L2)
//   img: (1,3,2048,2048) f32
//   out: [ lut (107811 floats) | result (1,3,2048,2048) ]
// Memory-bound: ~100MB HBM streaming + L2-resident gathers. No matrix math
// (per-lane interpolation weights -> WMMA inapplicable). CDNA5 paths used:
// async global<->LDS (ASYNCcnt), global_prefetch_b8, NT temporal hints.
// ---------------------------------------------------------------------------

typedef float __attribute__((ext_vector_type(4))) f32x4;   // native vector type
                                                           // (required by the
                                                           // nontemporal builtins)

#define LUT_S   33
#define LUT_S2  (33 * 33)          // 1089
#define LUT_S3  (33 * 33 * 33)     // 35937
#define LUT_N   (3 * LUT_S3)       // 107811 floats
#define NPIX    (2048 * 2048)      // 4194304 pixels per plane

// ---------------------------------------------------------------------------
// Kernel 1: LUT pass-through copy via the CDNA5 async global<->LDS path.
// Each lane moves 16B: global_load_async_to_lds_b128 -> s_wait_asynccnt ->
// global_store_async_from_lds_b128 (tracked on ASYNCcnt, ISA ch.10 / 15.18)
// ---------------------------------------------------------------------------
__global__ __launch_bounds__(256) void lut_copy_async(const float* __restrict__ lut,
                                                      float* __restrict__ out) {
    __shared__ f32x4 buf[256];
    constexpr int NV4 = LUT_N / 4;                 // 26952 full float4 chunks
    const int t = blockIdx.x * 256 + threadIdx.x;

    if (t < NV4) {
        // Low 32 bits of a generic shared-memory pointer are the LDS byte
        // address (ISA 10.2).
        unsigned lds_off = (unsigned)(size_t)(&buf[threadIdx.x]);
        unsigned long long src = (unsigned long long)(size_t)(lut + 4ull * t);
        unsigned long long dst = (unsigned long long)(size_t)(out + 4ull * t);

        asm volatile("global_load_async_to_lds_b128 %0, %1, off"
                     :: "v"(lds_off), "v"(src) : "memory");
        asm volatile("s_wait_asynccnt 0" ::: "memory");
        asm volatile("global_store_async_from_lds_b128 %0, %1, off"
                     :: "v"(dst), "v"(lds_off) : "memory");
        asm volatile("s_wait_asynccnt 0" ::: "memory");
    } else if (t == NV4) {
        // 107811 % 4 == 3 tail elements
        out[LUT_N - 3] = lut[LUT_N - 3];
        out[LUT_N - 2] = lut[LUT_N - 2];
        out[LUT_N - 1] = lut[LUT_N - 1];
    }
}

// ---------------------------------------------------------------------------
// Kernel 2: trilinear interpolation. 4 pixels / thread, NT b128 streaming,
// per-lane regular-temporal b32 gathers from the L2-resident LUT.
// ---------------------------------------------------------------------------
struct RGB { float r, g, b; };

__device__ __forceinline__ RGB trilerp1(const float* __restrict__ lut,
                                        float r, float g, float b) {
    // pix = clip(img * 32, 0, 32)   (algebraic collapse of the reference)
    float x = fminf(fmaxf(r * 32.0f, 0.0f), 32.0f);
    float y = fminf(fmaxf(g * 32.0f, 0.0f), 32.0f);
    float z = fminf(fmaxf(b * 32.0f, 0.0f), 32.0f);

    float xf = floorf(x), yf = floorf(y), zf = floorf(z);
    float fx = x - xf, fy = y - yf, fz = z - zf;
    int x0 = (int)xf, y0 = (int)yf, z0 = (int)zf;
    int x1 = min(x0 + 1, LUT_S - 1);
    int y1 = min(y0 + 1, LUT_S - 1);
    int z1 = min(z0 + 1, LUT_S - 1);

    int b00 = z0 * LUT_S2 + y0 * LUT_S;   // (z0, y0)
    int b01 = z0 * LUT_S2 + y1 * LUT_S;   // (z0, y1)
    int b10 = z1 * LUT_S2 + y0 * LUT_S;   // (z1, y0)
    int b11 = z1 * LUT_S2 + y1 * LUT_S;   // (z1, y1)

    float acc[3];
#pragma unroll
    for (int c = 0; c < 3; ++c) {
        const float* L = lut + c * LUT_S3;
        float c000 = L[b00 + x0], c001 = L[b00 + x1];
        float c010 = L[b01 + x0], c011 = L[b01 + x1];
        float c100 = L[b10 + x0], c101 = L[b10 + x1];
        float c110 = L[b11 + x0], c111 = L[b11 + x1];

        float c00 = fmaf(fx, c001 - c000, c000);
        float c01 = fmaf(fx, c011 - c010, c010);
        float c10 = fmaf(fx, c101 - c100, c100);
        float c11 = fmaf(fx, c111 - c110, c110);
        float c0  = fmaf(fy, c01 - c00, c00);
        float c1  = fmaf(fy, c11 - c10, c10);
        acc[c]    = fmaf(fz, c1 - c0, c0);
    }
    return RGB{acc[0], acc[1], acc[2]};
}

__global__ __launch_bounds__(256) void trilerp_kernel(const float* __restrict__ lut,
                                                      const float* __restrict__ img,
                                                      float* __restrict__ res) {
    const int t = blockIdx.x * 256 + threadIdx.x;   // 0 .. NPIX/4-1

    // Pull the whole 431KB LUT into L2 up-front (global_prefetch_b8):
    // one 128B cacheline per thread across the first ~13 blocks.
    constexpr int NLINES = (LUT_N + 31) / 32;       // 3370 cachelines
    if (t < NLINES) __builtin_prefetch(lut + t * 32, 0, 1);

    const int p = t * 4;
    // Streaming reads: touched exactly once -> non-temporal (th:TH_LOAD_NT),
    // keeps L2 dedicated to the hot LUT.
    const f32x4 r4 = __builtin_nontemporal_load((const f32x4*)(img + p));
    const f32x4 g4 = __builtin_nontemporal_load((const f32x4*)(img + NPIX + p));
    const f32x4 b4 = __builtin_nontemporal_load((const f32x4*)(img + 2 * NPIX + p));

    const RGB v0 = trilerp1(lut, r4.x, g4.x, b4.x);
    const RGB v1 = trilerp1(lut, r4.y, g4.y, b4.y);
    const RGB v2 = trilerp1(lut, r4.z, g4.z, b4.z);
    const RGB v3 = trilerp1(lut, r4.w, g4.w, b4.w);

    const f32x4 o0 = {v0.r, v1.r, v2.r, v3.r};
    const f32x4 o1 = {v0.g, v1.g, v2.g, v3.g};
    const f32x4 o2 = {v0.b, v1.b, v2.b, v3.b};

    // Streaming writes: never re-read -> non-temporal stores.
    __builtin_nontemporal_store(o0, (f32x4*)(res + p));
    __builtin_nontemporal_store(o1, (f32x4*)(res + NPIX + p));
    __builtin_nontemporal_store(o2, (f32x4*)(res + 2 * NPIX + p));
}

// ---------------------------------------------------------------------------
extern "C" void kernel_launch(void* const* d_in, const int* in_sizes, int n_in,
                              void* d_out, int out_size, void* d_ws, size_t ws_size,
                              hipStream_t stream) {
    const float* lut = (const float*)d_in[0];   // 107811 floats
    const float* img = (const float*)d_in[1];   // 12582912 floats
    float* out = (float*)d_out;

    // Output 0: LUT pass-through (async LDS copy path)
    constexpr int copy_threads = LUT_N / 4 + 1;             // 26953
    lut_copy_async<<<(copy_threads + 255) / 256, 256, 0, stream>>>(lut, out);

    // Output 1: trilinear-LUT result
    constexpr int nthreads = NPIX / 4;                      // 1048576
    trilerp_kernel<<<nthreads / 256, 256, 0, stream>>>(lut, img, out + LUT_N);
}